// CINBlock_88476326298189
// MI455X (gfx1250) — compile-verified
//
#include <hip/hip_runtime.h>

typedef __attribute__((ext_vector_type(2))) float v2f;
typedef __attribute__((ext_vector_type(4))) float v4f;
typedef __attribute__((ext_vector_type(8))) float v8f;
typedef __attribute__((ext_vector_type(4))) int   v4i;

#define GLOBAL_AS __attribute__((address_space(1)))
#define LDS_AS    __attribute__((address_space(3)))

#define CHUNK   128            // inner (i,j) columns staged per LDS buffer
#define LSTRIDE 132            // 128 + 4 pad -> conflict-free b64 column reads

#if __has_builtin(__builtin_amdgcn_global_load_async_to_lds_b128)
#define ASYNC_COPY 1
#else
#define ASYNC_COPY 0
#endif

__device__ __forceinline__ void stage_chunk(const float* __restrict__ W,
                                            int k0, int IJ, int c, float* dst) {
    // 16 rows x 128 cols; 256 threads * 32B each, fully coalesced per row.
    const int row = threadIdx.x >> 4;
    const int col = (threadIdx.x & 15) * 8;
    const float* src = W + (size_t)(k0 + row) * IJ + c * CHUNK + col;
    float* d = dst + row * LSTRIDE + col;
#if ASYNC_COPY
    __builtin_amdgcn_global_load_async_to_lds_b128(
        (GLOBAL_AS v4i*)src, (LDS_AS v4i*)d, 0, 0);
    __builtin_amdgcn_global_load_async_to_lds_b128(
        (GLOBAL_AS v4i*)(src + 4), (LDS_AS v4i*)(d + 4), 0, 0);
#else
    v4f v0 = *(const v4f*)src;
    v4f v1 = *(const v4f*)(src + 4);
    *(v4f*)d       = v0;
    *(v4f*)(d + 4) = v1;
#endif
}

__device__ __forceinline__ void wait_stage() {
#if ASYNC_COPY
#if __has_builtin(__builtin_amdgcn_s_wait_asynccnt)
    __builtin_amdgcn_s_wait_asynccnt(0);
#else
    asm volatile("s_wait_asynccnt 0x0" ::: "memory");
#endif
#endif
}

// One wave computes a full 16(k) x 32(d) output tile for one batch b.
// Block = 8 waves = 8 batches sharing the same k-tile of W staged in LDS.
__global__ __launch_bounds__(256) void cin_layer_kernel(
    const float* __restrict__ W,     // (256, Fh, 32) row-major
    const float* __restrict__ bias,  // (256)
    const float* __restrict__ h,     // (B, Fh, 32)
    const float* __restrict__ feat,  // (B, 32, 32)
    float* __restrict__ h_next,      // (B, 128, 32) (may be unused)
    float* __restrict__ out,         // (B, 512)
    int Fh, int out_off, int direct_k0, int write_h)
{
    const int IJ   = Fh * 32;
    const int k0   = blockIdx.x * 16;           // 16 k-tiles
    const int wave = threadIdx.x >> 5;
    const int lane = threadIdx.x & 31;
    const int b    = blockIdx.y * 8 + wave;     // 64 b-groups * 8 waves = 512
    const int m    = lane & 15;
    const int hh   = lane >> 4;                 // lane half (selects kk pair)

    __shared__ float Wlds[2][16 * LSTRIDE];     // double buffer, 16.9 KB total

    // Preload the feat columns this lane needs, indexed by compile-time s.
    // B-fragment element (s, v) needs feat[4s + 2*hh + v][d], d = m (+16 hi).
    const float* fb = feat + (size_t)b * (32 * 32);
    float g0_lo[8], g1_lo[8], g0_hi[8], g1_hi[8];
#pragma unroll
    for (int s = 0; s < 8; ++s) {
        int j0 = 4 * s + 2 * hh;
        g0_lo[s] = fb[j0 * 32 + m];
        g1_lo[s] = fb[(j0 + 1) * 32 + m];
        g0_hi[s] = fb[j0 * 32 + 16 + m];
        g1_hi[s] = fb[(j0 + 1) * 32 + 16 + m];
    }

    v8f acc_lo = {};
    v8f acc_hi = {};

    const float* hb = h + (size_t)b * IJ;
    const int nchunks = IJ / CHUNK;

    stage_chunk(W, k0, IJ, 0, &Wlds[0][0]);

    for (int c = 0; c < nchunks; ++c) {
        wait_stage();            // this wave's chunk c landed in LDS
        __syncthreads();         // all waves' chunk c landed; buf (c+1)&1 free
        if (c + 1 < nchunks)
            stage_chunk(W, k0, IJ, c + 1, &Wlds[(c + 1) & 1][0]);

        const float* wbuf = &Wlds[c & 1][0];
#pragma unroll
        for (int ii = 0; ii < 4; ++ii) {
            const int i = c * 4 + ii;
            const float h_lo = hb[i * 32 + m];
            const float h_hi = hb[i * 32 + 16 + m];
#pragma unroll
            for (int s = 0; s < 8; ++s) {
                // A fragment: W[k0+m][chunk cols], kk = 2*hh + {0,1}
                v2f a = *(const v2f*)&wbuf[m * LSTRIDE + ii * 32 + 4 * s + 2 * hh];
                v2f b_lo, b_hi;
                b_lo.x = h_lo * g0_lo[s];  b_lo.y = h_lo * g1_lo[s];
                b_hi.x = h_hi * g0_hi[s];  b_hi.y = h_hi * g1_hi[s];
                acc_lo = __builtin_amdgcn_wmma_f32_16x16x4_f32(
                    false, a, false, b_lo, (short)0, acc_lo, false, false);
                acc_hi = __builtin_amdgcn_wmma_f32_16x16x4_f32(
                    false, a, false, b_hi, (short)0, acc_hi, false, false);
            }
        }
    }

    // Epilogue: bias + relu; feed h_next; d-sum of direct rows -> out.
    float rsum[8];
#pragma unroll
    for (int r = 0; r < 8; ++r) {
        const int k = k0 + r + 8 * hh;          // output row for this VGPR/lane-half
        const float bv = bias[k];
        float lo = acc_lo[r] + bv;  lo = lo > 0.f ? lo : 0.f;
        float hi = acc_hi[r] + bv;  hi = hi > 0.f ? hi : 0.f;
        if (write_h && k < 128) {
            h_next[(size_t)b * (128 * 32) + k * 32 + m]      = lo;
            h_next[(size_t)b * (128 * 32) + k * 32 + 16 + m] = hi;
        }
        rsum[r] = lo + hi;                      // partial d-sum (d and d+16)
    }
#pragma unroll
    for (int r = 0; r < 8; ++r) {
        float v = rsum[r];
        v += __shfl_xor(v, 8);                  // masks <= 8 stay within 16-lane halves
        v += __shfl_xor(v, 4);
        v += __shfl_xor(v, 2);
        v += __shfl_xor(v, 1);
        const int k = k0 + r + 8 * hh;
        if (m == 0 && k >= direct_k0) {
            out[(size_t)b * 512 + out_off + (k - direct_k0)] = v;
        }
    }
}

extern "C" void kernel_launch(void* const* d_in, const int* in_sizes, int n_in,
                              void* d_out, int out_size, void* d_ws, size_t ws_size,
                              hipStream_t stream) {
    const float* feat = (const float*)d_in[0];
    const float* W0   = (const float*)d_in[1];
    const float* b0   = (const float*)d_in[2];
    const float* W1   = (const float*)d_in[3];
    const float* b1   = (const float*)d_in[4];
    const float* W2   = (const float*)d_in[5];
    const float* b2   = (const float*)d_in[6];
    float* out = (float*)d_out;

    float* h1 = (float*)d_ws;                    // (512,128,32)
    float* h2 = h1 + (size_t)512 * 128 * 32;     // (512,128,32)

    dim3 grid(16, 64), block(256);
    // layer 0: h = feat (Fh=32); direct rows 128..255 -> out cols [0,128)
    cin_layer_kernel<<<grid, block, 0, stream>>>(W0, b0, feat, feat, h1, out,
                                                 32, 0, 128, 1);
    // layer 1: h = h1 (Fh=128); direct rows 128..255 -> out cols [128,256)
    cin_layer_kernel<<<grid, block, 0, stream>>>(W1, b1, h1, feat, h2, out,
                                                 128, 128, 128, 1);
    // layer 2: h = h2 (Fh=128); all rows -> out cols [256,512)
    cin_layer_kernel<<<grid, block, 0, stream>>>(W2, b2, h2, feat, nullptr, out,
                                                 128, 256, 0, 0);
}